// DeStationaryAttention_42657615184247
// MI455X (gfx1250) — compile-verified
//
#include <hip/hip_runtime.h>
#include <hip/hip_bf16.h>
#include <math.h>

typedef float v2f __attribute__((ext_vector_type(2)));
typedef float v8f __attribute__((ext_vector_type(8)));
typedef int   v4i __attribute__((vector_size(16)));   // matches async-LDS builtin param

#define D_MODEL 512
#define N_HEADS 8
#define D_K     64
#define BATCH   4
#define SEQ     2048
#define BH      (BATCH * N_HEADS)   // 32
#define SROW    (SEQ + 4)           // 2052: 4-bank lane stride -> conflict-free frags
#define APAD    20                  // A-tile row pad: 80B rows (16B aligned), distinct banks

// ---------------------------------------------------------------------------
// Kernel 1: Y = X @ W + b, output remapped to [B*H, L, D_K] (head-major).
// Block = 128 threads (4 waves), block tile 64(M) x 64(N), K = 512.
// Tiles staged with GLOBAL_LOAD_ASYNC_TO_LDS_B128 (ASYNCcnt), math via f32 WMMA.
// ---------------------------------------------------------------------------
__global__ __launch_bounds__(128) void qkv_proj_kernel(
    const float* __restrict__ X,     // [B*L, 512]
    const float* __restrict__ W,     // [512, 512]
    const float* __restrict__ bias,  // [512]
    float* __restrict__ Yh)          // [B*H, L, 64]
{
  __shared__ float Asmem[64][APAD]; // 64x16 tile, rows padded to 80B
  __shared__ float Bsmem[16][80];   // 16x64 tile, rows padded to 320B

  const int tid   = threadIdx.x;
  const int wave  = tid >> 5;
  const int lane  = tid & 31;
  const int lhalf = lane & 15;
  const bool hi   = lane >= 16;

  const int mbase = blockIdx.x * 64;
  const int nbase = blockIdx.y * 64;

  // init accumulators with bias (same col for all 8 rows of a lane)
  v8f acc[4];
#pragma unroll
  for (int t = 0; t < 4; ++t) {
    float bv = bias[nbase + t * 16 + lhalf];
#pragma unroll
    for (int i = 0; i < 8; ++i) acc[t][i] = bv;
  }

  for (int kt = 0; kt < D_MODEL; kt += 16) {
    // async stage A: X[mbase..mbase+63][kt..kt+15], 256 x b128, 2 per thread
#pragma unroll
    for (int i = 0; i < 2; ++i) {
      int idx = tid + i * 128;            // 0..255
      int r   = idx >> 2;                 // 0..63
      int c4  = (idx & 3) * 4;            // 0,4,8,12
      __builtin_amdgcn_global_load_async_to_lds_b128(
          (v4i*)(X + (size_t)(mbase + r) * D_MODEL + kt + c4),
          (v4i*)&Asmem[r][c4], 0, 0);
    }
    // async stage B: W[kt..kt+15][nbase..nbase+63]
#pragma unroll
    for (int i = 0; i < 2; ++i) {
      int idx = tid + i * 128;
      int r   = idx >> 4;                 // 0..15
      int c4  = (idx & 15) * 4;           // 0..60
      __builtin_amdgcn_global_load_async_to_lds_b128(
          (v4i*)(W + (size_t)(kt + r) * D_MODEL + nbase + c4),
          (v4i*)&Bsmem[r][c4], 0, 0);
    }
    __builtin_amdgcn_s_wait_asynccnt(0);
    __syncthreads();

    const int mrow = wave * 16 + lhalf;   // A rows: both half-waves map M=0..15
#pragma unroll
    for (int kk = 0; kk < 16; kk += 4) {
      const int ks = kk + (hi ? 2 : 0);   // lanes 16-31 hold K=2,3
      v2f afrag;
      afrag.x = Asmem[mrow][ks];
      afrag.y = Asmem[mrow][ks + 1];
#pragma unroll
      for (int t = 0; t < 4; ++t) {
        v2f bfrag;
        bfrag.x = Bsmem[ks][t * 16 + lhalf];
        bfrag.y = Bsmem[ks + 1][t * 16 + lhalf];
        acc[t] = __builtin_amdgcn_wmma_f32_16x16x4_f32(
            false, afrag, false, bfrag, (short)0, acc[t], false, false);
      }
    }
    __syncthreads();
  }

  // store, remapped [B*L,512] -> [B*H, L, 64]
#pragma unroll
  for (int t = 0; t < 4; ++t) {
    int c = nbase + t * 16 + lhalf;
    int h = c >> 6, d = c & 63;
#pragma unroll
    for (int i = 0; i < 8; ++i) {
      int r = mbase + wave * 16 + i + (hi ? 8 : 0);
      int b = r >> 11, l = r & 2047;
      Yh[(size_t)((b * N_HEADS + h) * SEQ + l) * D_K + d] = acc[t][i];
    }
  }
}

// ---------------------------------------------------------------------------
// Kernel 2: attention for one (b,h, 16-query tile).
// 256 threads = 8 waves. Full 16x2048 score strip lives in LDS.
// ---------------------------------------------------------------------------
__global__ __launch_bounds__(256) void attn_kernel(
    const float* __restrict__ Q,   // [BH, L, 64]
    const float* __restrict__ K,   // [BH, L, 64]
    const float* __restrict__ V,   // [BH, L, 64]
    float* __restrict__ out,       // [B, L, 512]
    float* __restrict__ attw)      // [B, H, L, L]
{
  extern __shared__ float smem[];  // [16][SROW] scores + 4*256 reduce scratch

  const int tid   = threadIdx.x;
  const int wave  = tid >> 5;      // 0..7
  const int lane  = tid & 31;
  const int lhalf = lane & 15;
  const bool hi   = lane >= 16;

  const int qt    = blockIdx.x;    // 0..127
  const int bh    = blockIdx.y;    // 0..31
  const int qbase = qt * 16;

  const float* Qp = Q + (size_t)bh * SEQ * D_K;
  const float* Kp = K + (size_t)bh * SEQ * D_K;
  const float* Vp = V + (size_t)bh * SEQ * D_K;

  // ---- Phase 1: S = (Q/sqrt(dk)) @ K^T into LDS -------------------------
  // preload Q fragments for all 16 K-steps over d=0..63, pre-scaled by 1/8
  v2f qf[16];
  {
    const float* qrow = Qp + (size_t)(qbase + lhalf) * D_K + (hi ? 2 : 0);
#pragma unroll
    for (int s = 0; s < 16; ++s) {
      v2f v = *(const v2f*)(qrow + s * 4);
      qf[s].x = v.x * 0.125f;
      qf[s].y = v.y * 0.125f;
    }
  }
  for (int j = 0; j < 16; ++j) {
    const int nb = wave * 16 + j * 128;           // this wave's 16 key columns
    v8f acc = {0.f, 0.f, 0.f, 0.f, 0.f, 0.f, 0.f, 0.f};
    const float* krow = Kp + (size_t)(nb + lhalf) * D_K + (hi ? 2 : 0);
    __builtin_prefetch(krow + 128 * D_K, 0, 1);   // next j-tile of K
#pragma unroll
    for (int s = 0; s < 16; ++s) {
      v2f bf = *(const v2f*)(krow + s * 4);       // B[d][n] = K[n][d]
      acc = __builtin_amdgcn_wmma_f32_16x16x4_f32(
          false, qf[s], false, bf, (short)0, acc, false, false);
    }
#pragma unroll
    for (int i = 0; i < 8; ++i) {
      int r = i + (hi ? 8 : 0);
      smem[r * SROW + nb + lhalf] = acc[i];
    }
  }
  __syncthreads();

  // ---- Phase 2: row softmax; each wave owns rows {wave, wave+8} ---------
#pragma unroll
  for (int rr = 0; rr < 2; ++rr) {
    const int r = wave + rr * 8;
    float* Srow = smem + r * SROW;
    float m = -3.402823466e38f;
    for (int k = lane; k < SEQ; k += 32) m = fmaxf(m, Srow[k]);
#pragma unroll
    for (int off = 16; off > 0; off >>= 1) m = fmaxf(m, __shfl_xor(m, off, 32));
    float ssum = 0.f;
    for (int k = lane; k < SEQ; k += 32) {
      float e = __expf(Srow[k] - m);
      Srow[k] = e;
      ssum += e;
    }
#pragma unroll
    for (int off = 16; off > 0; off >>= 1) ssum += __shfl_xor(ssum, off, 32);
    const float inv = 1.0f / ssum;
    float* arow = attw + ((size_t)bh * SEQ + qbase + r) * SEQ;
    for (int k = lane; k < SEQ; k += 32) {
      float p = Srow[k] * inv;
      Srow[k] = p;          // keep normalized P in LDS for phase 3
      arow[k] = p;          // coalesced stream to attn_weights output
    }
  }
  __syncthreads();

  // ---- Phase 3: O = P @ V ----------------------------------------------
  // wave -> (d-tile = wave&3, k-half = wave>>2); partials merged via LDS.
  const int dtile = (wave & 3) * 16;
  const int kbeg  = (wave >> 2) * 1024;
  v8f oacc = {0.f, 0.f, 0.f, 0.f, 0.f, 0.f, 0.f, 0.f};
  const float* prow = smem + lhalf * SROW;        // A rows = query rows 0..15
  for (int kk = kbeg; kk < kbeg + 1024; kk += 4) {
    const int ks = kk + (hi ? 2 : 0);
    v2f af = *(const v2f*)(prow + ks);            // P[row][ks], P[row][ks+1]
    v2f bf;
    const float* vcol = Vp + (size_t)ks * D_K + dtile + lhalf;
    __builtin_prefetch(vcol + 16 * D_K, 0, 1);    // V a few k-rows ahead
    bf.x = vcol[0];                               // V[ks]  [d]
    bf.y = vcol[D_K];                             // V[ks+1][d]
    oacc = __builtin_amdgcn_wmma_f32_16x16x4_f32(
        false, af, false, bf, (short)0, oacc, false, false);
  }

  float* scratch = smem + 16 * SROW;              // 4 tiles * 256 floats
  if (wave >= 4) {
    float* dst = scratch + (wave & 3) * 256 + (hi ? 128 : 0) + lhalf;
#pragma unroll
    for (int i = 0; i < 8; ++i) dst[i * 16] = oacc[i];
  }
  __syncthreads();
  if (wave < 4) {
    const float* src = scratch + (wave & 3) * 256 + (hi ? 128 : 0) + lhalf;
    const int b = bh >> 3, h = bh & 7;
#pragma unroll
    for (int i = 0; i < 8; ++i) {
      float val = oacc[i] + src[i * 16];
      int r = qbase + i + (hi ? 8 : 0);
      out[((size_t)(b * SEQ + r)) * D_MODEL + h * D_K + dtile + lhalf] = val;
    }
  }
}

// ---------------------------------------------------------------------------
extern "C" void kernel_launch(void* const* d_in, const int* in_sizes, int n_in,
                              void* d_out, int out_size, void* d_ws, size_t ws_size,
                              hipStream_t stream) {
  (void)in_sizes; (void)n_in; (void)out_size; (void)ws_size;
  const float* q_in = (const float*)d_in[0];   // [B, L, 512]
  const float* k_in = (const float*)d_in[1];
  const float* v_in = (const float*)d_in[2];
  const float* Wq   = (const float*)d_in[3];
  const float* bq   = (const float*)d_in[4];
  const float* Wk   = (const float*)d_in[5];
  const float* bk   = (const float*)d_in[6];
  const float* Wv   = (const float*)d_in[7];
  const float* bv   = (const float*)d_in[8];

  float* out  = (float*)d_out;                          // [B, L, 512]
  float* attw = out + (size_t)BATCH * SEQ * D_MODEL;    // [B, H, L, L]

  // workspace: projected Q, K, V in [B*H, L, 64] layout (3 * 16 MB)
  float* Qh = (float*)d_ws;
  float* Kh = Qh + (size_t)BH * SEQ * D_K;
  float* Vh = Kh + (size_t)BH * SEQ * D_K;

  dim3 pgrid((BATCH * SEQ) / 64, D_MODEL / 64);         // 128 x 8
  qkv_proj_kernel<<<pgrid, 128, 0, stream>>>(q_in, Wq, bq, Qh);
  qkv_proj_kernel<<<pgrid, 128, 0, stream>>>(k_in, Wk, bk, Kh);
  qkv_proj_kernel<<<pgrid, 128, 0, stream>>>(v_in, Wv, bv, Vh);

  const size_t shbytes = (size_t)(16 * SROW + 4 * 256) * sizeof(float);
  dim3 agrid(SEQ / 16, BH);                             // 128 x 32
  attn_kernel<<<agrid, 256, shbytes, stream>>>(Qh, Kh, Vh, out, attw);
}